// APPNP_Model_5342939316768
// MI455X (gfx1250) — compile-verified
//
#include <hip/hip_runtime.h>

// ---------------------------------------------------------------------------
// APPNP GCN for MI455X (gfx1250, wave32).
//   Phase 1: fused 2-layer MLP with V_WMMA_F32_16X16X4_F32 (fp32 matrix core)
//   Phase 2: degree/norm setup
//   Phase 3: K=10 propagation steps (combine-init + edge scatter-add),
//            ping-pong between d_ws buffer and d_out; whole set lives in L2.
// ---------------------------------------------------------------------------

typedef __attribute__((ext_vector_type(2))) float v2f;
typedef __attribute__((ext_vector_type(8))) float v8f;

#define IN_C  512
#define HID_C 256
#define OUT_C 64
#define KITER 10
#define LDS_STRIDE (HID_C + 4)   // pad to avoid LDS bank conflicts

// ---------------------------------------------------------------------------
// Fused MLP: h0 = relu(x @ w1.T + b1) @ w2.T + b2
// One wave per 16-row strip, 2 waves (64 threads) per block.
// GEMM1: 16 tiles of 16x16 (full 256 hid channels) accumulated in VGPRs.
// GEMM2: via LDS round-trip, 4 tiles of 16x16 (64 out channels).
// ---------------------------------------------------------------------------
__global__ __launch_bounds__(64)
void mlp_wmma_kernel(const float* __restrict__ x,
                     const float* __restrict__ w1,
                     const float* __restrict__ b1,
                     const float* __restrict__ w2,
                     const float* __restrict__ b2,
                     float* __restrict__ h0,
                     float* __restrict__ hcur,
                     int nodes)
{
    __shared__ float lds[2][16][LDS_STRIDE];

    const int wave = threadIdx.x >> 5;
    const int lane = threadIdx.x & 31;
    const int m0   = (blockIdx.x * 2 + wave) * 16;
    if (m0 >= nodes) return;              // wave-uniform (never splits EXEC)

    const int half = lane >> 4;           // 0: K pair {0,1}, 1: K pair {2,3}
    const int l15  = lane & 15;

    // ---------------- GEMM1: hid[16][256] = x_strip @ w1.T -----------------
    v8f acc[16];
    #pragma unroll
    for (int t = 0; t < 16; ++t) acc[t] = (v8f){};

    const float* xrow = x + (size_t)(m0 + l15) * IN_C + 2 * half;
    for (int k0 = 0; k0 < IN_C; k0 += 4) {
        v2f a = *(const v2f*)(xrow + k0);
        #pragma unroll
        for (int t = 0; t < 16; ++t) {
            // B[k][n] = w1[n][k], n = t*16 + l15 -> contiguous float2 in w1 row
            v2f b = *(const v2f*)(w1 + (size_t)(t * 16 + l15) * IN_C + k0 + 2 * half);
            acc[t] = __builtin_amdgcn_wmma_f32_16x16x4_f32(
                false, a, false, b, (short)0, acc[t], false, false);
        }
    }

    // bias + relu -> LDS (C layout: VGPR v, lane -> row v+8*half, col t*16+l15)
    #pragma unroll
    for (int t = 0; t < 16; ++t) {
        float bias = b1[t * 16 + l15];
        #pragma unroll
        for (int v = 0; v < 8; ++v) {
            float val = acc[t][v] + bias;
            lds[wave][v + 8 * half][t * 16 + l15] = fmaxf(val, 0.0f);
        }
    }
    __syncthreads();

    // ---------------- GEMM2: out[16][64] = hid @ w2.T ----------------------
    v8f acc2[4];
    #pragma unroll
    for (int t = 0; t < 4; ++t) acc2[t] = (v8f){};

    for (int k0 = 0; k0 < HID_C; k0 += 4) {
        v2f a = *(const v2f*)(&lds[wave][l15][k0 + 2 * half]);
        #pragma unroll
        for (int t = 0; t < 4; ++t) {
            v2f b = *(const v2f*)(w2 + (size_t)(t * 16 + l15) * HID_C + k0 + 2 * half);
            acc2[t] = __builtin_amdgcn_wmma_f32_16x16x4_f32(
                false, a, false, b, (short)0, acc2[t], false, false);
        }
    }

    // epilogue: add b2, write h0 and initial h_cur (= d_out)
    #pragma unroll
    for (int t = 0; t < 4; ++t) {
        float bias = b2[t * 16 + l15];
        #pragma unroll
        for (int v = 0; v < 8; ++v) {
            int row = m0 + v + 8 * half;
            int col = t * 16 + l15;
            float val = acc2[t][v] + bias;
            h0[(size_t)row * OUT_C + col]   = val;
            hcur[(size_t)row * OUT_C + col] = val;
        }
    }
}

// ---------------------------------------------------------------------------
// Normalization setup
// ---------------------------------------------------------------------------
__global__ __launch_bounds__(256)
void deg_init_kernel(float* __restrict__ deg, int n) {
    int i = blockIdx.x * blockDim.x + threadIdx.x;
    if (i < n) deg[i] = 1.0f;                       // self-loop
}

__global__ __launch_bounds__(256)
void deg_scatter_kernel(const int* __restrict__ dst, float* __restrict__ deg, int e) {
    int i = blockIdx.x * blockDim.x + threadIdx.x;
    if (i < e) atomicAdd(&deg[dst[i]], 1.0f);
}

__global__ __launch_bounds__(256)
void rsqrt_kernel(float* __restrict__ deg, int n) {
    int i = blockIdx.x * blockDim.x + threadIdx.x;
    if (i < n) {
        float d = deg[i];
        deg[i] = (d > 0.0f) ? rsqrtf(d) : 0.0f;     // deg -> dinv in place
    }
}

__global__ __launch_bounds__(256)
void wnorm_kernel(const int* __restrict__ src, const int* __restrict__ dst,
                  const float* __restrict__ dinv, float* __restrict__ wnorm, int e) {
    int i = blockIdx.x * blockDim.x + threadIdx.x;
    if (i < e) wnorm[i] = 0.9f * dinv[src[i]] * dinv[dst[i]];  // (1-alpha)*norm
}

// ---------------------------------------------------------------------------
// Propagation step part 1: h_next = alpha*h0 + (1-alpha)*selfnorm*h_cur
// (covers the self-loop message and the teleport term; float4 vectorized)
// ---------------------------------------------------------------------------
__global__ __launch_bounds__(256)
void combine_init_kernel(const float* __restrict__ h0,
                         const float* __restrict__ hcur,
                         const float* __restrict__ dinv,
                         float* __restrict__ hnext, int n) {
    int i = blockIdx.x * blockDim.x + threadIdx.x;   // over n*16 float4 groups
    if (i >= n * (OUT_C / 4)) return;
    int node = i / (OUT_C / 4);
    float di = dinv[node];
    float ws = 0.9f * di * di;
    float4 a = ((const float4*)h0)[i];
    float4 b = ((const float4*)hcur)[i];
    float4 r;
    r.x = 0.1f * a.x + ws * b.x;
    r.y = 0.1f * a.y + ws * b.y;
    r.z = 0.1f * a.z + ws * b.z;
    r.w = 0.1f * a.w + ws * b.w;
    ((float4*)hnext)[i] = r;
}

// ---------------------------------------------------------------------------
// Propagation step part 2: edge scatter-add. One wave per edge, 64 channels
// as float2 per lane; f32 atomics resolve in L2 (h_next is L2-resident).
// ---------------------------------------------------------------------------
__global__ __launch_bounds__(256)
void edge_scatter_kernel(const int* __restrict__ src, const int* __restrict__ dst,
                         const float* __restrict__ wnorm,
                         const float* __restrict__ hcur,
                         float* __restrict__ hnext, int e) {
    int edge = blockIdx.x * (blockDim.x >> 5) + (threadIdx.x >> 5);
    if (edge >= e) return;
    int lane = threadIdx.x & 31;
    int s = src[edge];
    int d = dst[edge];
    float w = wnorm[edge];
    float2 v = *(const float2*)(hcur + (size_t)s * OUT_C + lane * 2);
    float* p = hnext + (size_t)d * OUT_C + lane * 2;
    atomicAdd(p + 0, w * v.x);
    atomicAdd(p + 1, w * v.y);
}

// ---------------------------------------------------------------------------
extern "C" void kernel_launch(void* const* d_in, const int* in_sizes, int n_in,
                              void* d_out, int out_size, void* d_ws, size_t ws_size,
                              hipStream_t stream) {
    const float* x  = (const float*)d_in[0];
    const int*   ei = (const int*)  d_in[1];
    const float* w1 = (const float*)d_in[2];
    const float* b1 = (const float*)d_in[3];
    const float* w2 = (const float*)d_in[4];
    const float* b2 = (const float*)d_in[5];
    float* out = (float*)d_out;

    const int nodes = in_sizes[0] / IN_C;     // 100000
    const int edges = in_sizes[1] / 2;        // 1600000
    const int* src = ei;
    const int* dst = ei + edges;

    // workspace layout (floats): h0 | bufA | dinv | wnorm   (~58 MB total)
    float* h0    = (float*)d_ws;
    float* bufA  = h0   + (size_t)nodes * OUT_C;
    float* dinv  = bufA + (size_t)nodes * OUT_C;
    float* wnorm = dinv + nodes;

    // Phase 1: fused WMMA MLP -> h0 (and initial h_cur in d_out)
    int strips = (nodes + 15) / 16;
    mlp_wmma_kernel<<<(strips + 1) / 2, 64, 0, stream>>>(x, w1, b1, w2, b2,
                                                         h0, out, nodes);

    // Phase 2: symmetric GCN normalization
    deg_init_kernel   <<<(nodes + 255) / 256, 256, 0, stream>>>(dinv, nodes);
    deg_scatter_kernel<<<(edges + 255) / 256, 256, 0, stream>>>(dst, dinv, edges);
    rsqrt_kernel      <<<(nodes + 255) / 256, 256, 0, stream>>>(dinv, nodes);
    wnorm_kernel      <<<(edges + 255) / 256, 256, 0, stream>>>(src, dst, dinv,
                                                                wnorm, edges);

    // Phase 3: K propagation steps, ping-pong so step 10 lands in d_out
    float* hcur  = out;
    float* hnext = bufA;
    for (int it = 0; it < KITER; ++it) {
        int nvec = nodes * (OUT_C / 4);
        combine_init_kernel<<<(nvec + 255) / 256, 256, 0, stream>>>(h0, hcur,
                                                                    dinv, hnext,
                                                                    nodes);
        edge_scatter_kernel<<<(edges + 7) / 8, 256, 0, stream>>>(src, dst, wnorm,
                                                                 hcur, hnext,
                                                                 edges);
        float* tmp = hcur; hcur = hnext; hnext = tmp;
    }
    // after an even number of swaps, hcur == out : final result is in d_out
}